// Attention_71665824301254
// MI455X (gfx1250) — compile-verified
//
#include <hip/hip_runtime.h>
#include <hip/hip_bf16.h>

// Problem constants (from reference)
constexpr int kN  = 50000;
constexpr int kE  = 600000;
constexpr int kD  = 128;
constexpr int kH  = 2;
constexpr int kC  = 64;   // per-head channels
constexpr int kED = 5;

typedef float v2f __attribute__((ext_vector_type(2)));
typedef float v8f __attribute__((ext_vector_type(8)));

// ---------------------------------------------------------------------------
// Generic fill
// ---------------------------------------------------------------------------
__global__ __launch_bounds__(256) void fill_kernel(float* __restrict__ p, float v, int n) {
    int i = blockIdx.x * 256 + threadIdx.x;
    if (i < n) p[i] = v;
}

// ---------------------------------------------------------------------------
// out[16-row tile] = A @ W^T + bias, D=128, fp32 WMMA (16x16x4)
// grid.x = N/16, block = 256 (8 waves); wave w owns N-tile w (cols 16w..16w+15)
// ---------------------------------------------------------------------------
__global__ __launch_bounds__(256) void gemm128_kernel(
    const float* __restrict__ A, const float* __restrict__ W,
    const float* __restrict__ bias, float* __restrict__ out)
{
    __shared__ float sA[16 * 132];                 // padded stride: no bank conflicts
    const float* Ab = A + (size_t)blockIdx.x * 16 * kD;
    for (int i = threadIdx.x; i < 16 * kD; i += 256)
        sA[(i >> 7) * 132 + (i & 127)] = Ab[i];
    __syncthreads();

    const int wave  = threadIdx.x >> 5;            // 0..7 -> N tile
    const int lane  = threadIdx.x & 31;
    const int lm    = lane & 15;
    const int khalf = (lane >> 4) * 2;             // K sub-pair: 0 or 2
    const int n     = wave * 16 + lm;              // global output column

    v8f acc;
    const float bval = bias[n];
#pragma unroll
    for (int i = 0; i < 8; ++i) acc[i] = bval;

    const float* Wn = W + (size_t)n * kD;          // W row n => B[k][n] = W[n][k]
#pragma unroll
    for (int kk = 0; kk < kD; kk += 4) {
        const float2 av = *(const float2*)&sA[lm * 132 + kk + khalf]; // A[m][k..k+1]
        const float2 bv = *(const float2*)&Wn[kk + khalf];            // B[k..k+1][n]
        v2f a; a[0] = av.x; a[1] = av.y;
        v2f b; b[0] = bv.x; b[1] = bv.y;
        acc = __builtin_amdgcn_wmma_f32_16x16x4_f32(false, a, false, b,
                                                    (short)0, acc, false, false);
    }

    float* outb = out + (size_t)blockIdx.x * 16 * kD;
    const int row0 = (lane >> 4) * 8;              // C/D layout: row = r + 8*(lane/16)
#pragma unroll
    for (int r = 0; r < 8; ++r)
        outb[(row0 + r) * kD + n] = acc[r];
}

// ---------------------------------------------------------------------------
// Pass 1: per-edge alpha[e][h] = dot(q[dst], k[src] + edge_attr@We^T)/sqrt(C)
//         + atomic segment max into amax[dst][h].  One edge per wave32.
// ---------------------------------------------------------------------------
__global__ __launch_bounds__(256) void edge_alpha_kernel(
    const float* __restrict__ q, const float* __restrict__ k,
    const int* __restrict__ ei, const float* __restrict__ eattr,
    const float* __restrict__ We, float* __restrict__ alpha,
    float* __restrict__ amax)
{
    __shared__ float sWe[kD * kED];                // 640 floats = 2.5 KB
    for (int i = threadIdx.x; i < kD * kED; i += 256) sWe[i] = We[i];
    __syncthreads();

    const int wave = threadIdx.x >> 5;
    const int lane = threadIdx.x & 31;
    const int e    = blockIdx.x * 8 + wave;
    if (e >= kE) return;

    const int src = ei[e];
    const int dst = ei[kE + e];
    const float ea0 = eattr[e * kED + 0], ea1 = eattr[e * kED + 1],
                ea2 = eattr[e * kED + 2], ea3 = eattr[e * kED + 3],
                ea4 = eattr[e * kED + 4];

    const int c4 = lane * 4;                       // lanes 0-15: head 0, 16-31: head 1
    const float4 qv = *(const float4*)(q + (size_t)dst * kD + c4);
    const float4 kv = *(const float4*)(k + (size_t)src * kD + c4);
    float qa[4] = {qv.x, qv.y, qv.z, qv.w};
    float ka[4] = {kv.x, kv.y, kv.z, kv.w};

    float p = 0.0f;
#pragma unroll
    for (int cc = 0; cc < 4; ++cc) {
        const int c = c4 + cc;
        const float ec = ea0 * sWe[c * kED + 0] + ea1 * sWe[c * kED + 1] +
                         ea2 * sWe[c * kED + 2] + ea3 * sWe[c * kED + 3] +
                         ea4 * sWe[c * kED + 4];
        p += qa[cc] * (ka[cc] + ec);
    }
    // 16-lane tree: per-head dot (xor masks < 16 never cross the head boundary)
    p += __shfl_xor(p, 8);
    p += __shfl_xor(p, 4);
    p += __shfl_xor(p, 2);
    p += __shfl_xor(p, 1);

    const float aval = p * 0.125f;                 // 1/sqrt(64)
    if ((lane & 15) == 0) {
        const int h = lane >> 4;
        alpha[(size_t)e * kH + h] = aval;
        unsafeAtomicMax(&amax[(size_t)dst * kH + h], aval);  // global_atomic_max_num_f32
    }
}

// ---------------------------------------------------------------------------
// Pass 2: ex = exp(alpha - amax[dst]); alpha <- ex; denom[dst] += ex
// ---------------------------------------------------------------------------
__global__ __launch_bounds__(256) void edge_softmax_kernel(
    const int* __restrict__ ei, const float* __restrict__ amax,
    float* __restrict__ alpha, float* __restrict__ denom)
{
    const int tid = blockIdx.x * 256 + threadIdx.x;
    if (tid >= kE * kH) return;
    const int e = tid >> 1, h = tid & 1;
    const int dst = ei[kE + e];
    const float ex = __expf(alpha[tid] - amax[(size_t)dst * kH + h]);
    alpha[tid] = ex;
    unsafeAtomicAdd(&denom[(size_t)dst * kH + h], ex);       // global_atomic_add_f32
}

// ---------------------------------------------------------------------------
// Pass 3: out_agg[dst] += (ex/denom[dst]) * (v[src] + edge_attr@We^T)
// One edge per wave32, 4 channels/lane, native fp32 scatter atomics.
// ---------------------------------------------------------------------------
__global__ __launch_bounds__(256) void edge_scatter_kernel(
    const float* __restrict__ v, const int* __restrict__ ei,
    const float* __restrict__ eattr, const float* __restrict__ We,
    const float* __restrict__ alpha, const float* __restrict__ denom,
    float* __restrict__ out_agg)
{
    __shared__ float sWe[kD * kED];
    for (int i = threadIdx.x; i < kD * kED; i += 256) sWe[i] = We[i];
    __syncthreads();

    const int wave = threadIdx.x >> 5;
    const int lane = threadIdx.x & 31;
    const int e    = blockIdx.x * 8 + wave;
    if (e >= kE) return;

    const int src = ei[e];
    const int dst = ei[kE + e];
    const int h   = lane >> 4;
    const float wgt = alpha[(size_t)e * kH + h] / denom[(size_t)dst * kH + h];

    const float ea0 = eattr[e * kED + 0], ea1 = eattr[e * kED + 1],
                ea2 = eattr[e * kED + 2], ea3 = eattr[e * kED + 3],
                ea4 = eattr[e * kED + 4];

    const int c4 = lane * 4;
    const float4 vv = *(const float4*)(v + (size_t)src * kD + c4);
    float va[4] = {vv.x, vv.y, vv.z, vv.w};
    float* outb = out_agg + (size_t)dst * kD + c4;
#pragma unroll
    for (int cc = 0; cc < 4; ++cc) {
        const int c = c4 + cc;
        const float ec = ea0 * sWe[c * kED + 0] + ea1 * sWe[c * kED + 1] +
                         ea2 * sWe[c * kED + 2] + ea3 * sWe[c * kED + 3] +
                         ea4 * sWe[c * kED + 4];
        unsafeAtomicAdd(&outb[cc], wgt * (va[cc] + ec));
    }
}

// ---------------------------------------------------------------------------
// beta gate: beta = sigmoid([out, x_r, out-x_r] @ Wbeta^T);
//            gated = beta*x_r + (1-beta)*out.  One node per wave32.
// ---------------------------------------------------------------------------
__global__ __launch_bounds__(256) void beta_gate_kernel(
    const float* __restrict__ out_agg, const float* __restrict__ xr,
    const float* __restrict__ Wbeta, float* __restrict__ gated)
{
    __shared__ float sWb[3 * kD];                  // 384 floats
    for (int i = threadIdx.x; i < 3 * kD; i += 256) sWb[i] = Wbeta[i];
    __syncthreads();

    const int wave = threadIdx.x >> 5;
    const int lane = threadIdx.x & 31;
    const int node = blockIdx.x * 8 + wave;
    if (node >= kN) return;

    const int c4 = lane * 4;
    const float4 ov = *(const float4*)(out_agg + (size_t)node * kD + c4);
    const float4 xv = *(const float4*)(xr + (size_t)node * kD + c4);
    float oa[4] = {ov.x, ov.y, ov.z, ov.w};
    float xa[4] = {xv.x, xv.y, xv.z, xv.w};

    float p = 0.0f;
#pragma unroll
    for (int cc = 0; cc < 4; ++cc) {
        const int c = c4 + cc;
        p += sWb[c] * oa[cc] + sWb[kD + c] * xa[cc] + sWb[2 * kD + c] * (oa[cc] - xa[cc]);
    }
    p += __shfl_xor(p, 16);
    p += __shfl_xor(p, 8);
    p += __shfl_xor(p, 4);
    p += __shfl_xor(p, 2);
    p += __shfl_xor(p, 1);

    const float beta = 1.0f / (1.0f + __expf(-p));
    float4 g;
    g.x = beta * xa[0] + (1.0f - beta) * oa[0];
    g.y = beta * xa[1] + (1.0f - beta) * oa[1];
    g.z = beta * xa[2] + (1.0f - beta) * oa[2];
    g.w = beta * xa[3] + (1.0f - beta) * oa[3];
    *(float4*)(gated + (size_t)node * kD + c4) = g;
}

// ---------------------------------------------------------------------------
extern "C" void kernel_launch(void* const* d_in, const int* in_sizes, int n_in,
                              void* d_out, int out_size, void* d_ws, size_t ws_size,
                              hipStream_t stream)
{
    const float* x     = (const float*)d_in[0];
    const int*   ei    = (const int*)  d_in[1];
    const float* eattr = (const float*)d_in[2];
    const float* Wq    = (const float*)d_in[3];
    const float* bq    = (const float*)d_in[4];
    const float* Wk    = (const float*)d_in[5];
    const float* bk    = (const float*)d_in[6];
    const float* Wv    = (const float*)d_in[7];
    const float* bv    = (const float*)d_in[8];
    const float* We    = (const float*)d_in[9];
    const float* Wskip = (const float*)d_in[10];
    const float* bskip = (const float*)d_in[11];
    const float* Wbeta = (const float*)d_in[12];
    const float* Wproj = (const float*)d_in[13];
    const float* bproj = (const float*)d_in[14];

    float* ws = (float*)d_ws;
    const size_t ND = (size_t)kN * kD;
    float* q       = ws;                // reused as 'gated' after pass 1 (q dead then)
    float* kb      = ws + 1 * ND;
    float* vb      = ws + 2 * ND;
    float* xr      = ws + 3 * ND;
    float* out_agg = ws + 4 * ND;
    float* alpha   = ws + 5 * ND;
    float* amax    = alpha + (size_t)kE * kH;
    float* denom   = amax + (size_t)kN * kH;
    float* gated   = q;

    const int rowTiles = kN / 16;       // 3125

    // node projections (fp32 WMMA GEMMs)
    gemm128_kernel<<<rowTiles, 256, 0, stream>>>(x, Wq, bq, q);
    gemm128_kernel<<<rowTiles, 256, 0, stream>>>(x, Wk, bk, kb);
    gemm128_kernel<<<rowTiles, 256, 0, stream>>>(x, Wv, bv, vb);
    gemm128_kernel<<<rowTiles, 256, 0, stream>>>(x, Wskip, bskip, xr);

    // init segment buffers
    fill_kernel<<<(kN * kH + 255) / 256, 256, 0, stream>>>(amax, -3.0e38f, kN * kH);
    fill_kernel<<<(kN * kH + 255) / 256, 256, 0, stream>>>(denom, 0.0f, kN * kH);
    fill_kernel<<<((int)ND + 255) / 256, 256, 0, stream>>>(out_agg, 0.0f, (int)ND);

    // edge passes
    const int edgeBlocks = (kE + 7) / 8;   // 8 edges (waves) per block
    edge_alpha_kernel<<<edgeBlocks, 256, 0, stream>>>(q, kb, ei, eattr, We, alpha, amax);
    edge_softmax_kernel<<<(kE * kH + 255) / 256, 256, 0, stream>>>(ei, amax, alpha, denom);
    edge_scatter_kernel<<<edgeBlocks, 256, 0, stream>>>(vb, ei, eattr, We, alpha, denom, out_agg);

    // beta-gated skip, then output projection
    beta_gate_kernel<<<(kN + 7) / 8, 256, 0, stream>>>(out_agg, xr, Wbeta, gated);
    gemm128_kernel<<<rowTiles, 256, 0, stream>>>(gated, Wproj, bproj, (float*)d_out);
}